// EGATConv_14817637171461
// MI455X (gfx1250) — compile-verified
//
#include <hip/hip_runtime.h>
#include <hip/hip_bf16.h>
#include <math.h>

// Problem constants (from reference)
#define NN     100000
#define EE     1600000
#define IN_N   128
#define IN_E   64
#define HH     4
#define FN_    32
#define FE_    16
#define HFN    (HH*FN_)   // 128
#define HFE    (HH*FE_)   // 64
#define NTILES (NN/16)    // 6250  (exact)
#define ETILES (EE/16)    // 100000 (exact)

typedef __attribute__((ext_vector_type(2))) float v2f;
typedef __attribute__((ext_vector_type(8))) float v8f;

__device__ __forceinline__ v8f wmma_f32_16x16x4(v2f a, v2f b, v8f c) {
#if defined(__gfx1250__)
  // 8 args: (neg_a, A, neg_b, B, c_mod, C, reuse_a, reuse_b)
  return __builtin_amdgcn_wmma_f32_16x16x4_f32(false, a, false, b, (short)0, c,
                                               false, false);
#else
  (void)a; (void)b;
  return c;
#endif
}

__device__ __forceinline__ void atomicMaxFloat(float* addr, float val) {
  // int-trick float atomic max (correct for all sign combos, init = -inf)
  if (val >= 0.0f)
    atomicMax((int*)addr, __float_as_int(val));
  else
    atomicMin((unsigned int*)addr, __float_as_uint(val));
}

// ---------------------------------------------------------------------------
// Kernel 1: re-initialize accumulation buffers every call (graph replay safe)
// ---------------------------------------------------------------------------
__global__ void egat_init(float* h_out, float* seg_max, float* seg_sum) {
  long idx = (long)blockIdx.x * blockDim.x + threadIdx.x;
  if (idx < (long)NN * HFN) h_out[idx] = 0.0f;
  if (idx < (long)NN * HH) {
    seg_max[idx] = -INFINITY;
    seg_sum[idx] = 0.0f;
  }
}

// ---------------------------------------------------------------------------
// Kernel 2: node projections with V_WMMA_F32_16X16X4_F32
//   f_ni = nfeats @ W_ni            [N, 64]
//   f_nj = nfeats @ W_nj            [N, 64]
//   h    = nfeats @ W_node + b_node [N, 128]
// One wave per (16-row tile, 64-col slab). blockIdx.y selects slab:
//   0 -> W_ni cols 0..63, 1 -> W_nj cols 0..63,
//   2 -> W_node cols 0..63, 3 -> W_node cols 64..127
// Loads for a k-step are batched (A + 4 B pairs) before the 4 WMMAs so the
// scheduler can cover them with one wait and co-execute the WMMA stream.
// ---------------------------------------------------------------------------
__global__ __launch_bounds__(32)
void egat_node_proj(const float* __restrict__ nf,
                    const float* __restrict__ Wni,
                    const float* __restrict__ Wnj,
                    const float* __restrict__ Wnode,
                    const float* __restrict__ bnode,
                    float* __restrict__ f_ni,
                    float* __restrict__ f_nj,
                    float* __restrict__ h) {
  const int lane    = threadIdx.x;
  const int rowTile = blockIdx.x;
  const int sel     = blockIdx.y;

  const float* W;
  float*       out;
  int          outW, colbase;
  bool         addb = false;
  if (sel == 0)      { W = Wni;   out = f_ni; outW = HFE; colbase = 0;  }
  else if (sel == 1) { W = Wnj;   out = f_nj; outW = HFE; colbase = 0;  }
  else if (sel == 2) { W = Wnode; out = h;    outW = HFN; colbase = 0;  addb = true; }
  else               { W = Wnode; out = h;    outW = HFN; colbase = 64; addb = true; }

  const int n16   = lane & 15;
  const int kboff = (lane >> 4) * 2;         // K-pair select per ISA A layout
  const int mrow  = rowTile * 16 + n16;      // A row for this lane

  v8f acc[4];
#pragma unroll
  for (int t = 0; t < 4; ++t)
#pragma unroll
    for (int v = 0; v < 8; ++v) acc[t][v] = 0.0f;

  const float* arow = nf + (long)mrow * IN_N;
  for (int k = 0; k < IN_N; k += 4) {
    v2f a;
    a.x = arow[k + kboff];
    a.y = arow[k + kboff + 1];
    v2f b[4];
#pragma unroll
    for (int t = 0; t < 4; ++t) {
      const int n = colbase + t * 16 + n16;
      b[t].x = W[(long)(k + kboff)     * outW + n];
      b[t].y = W[(long)(k + kboff + 1) * outW + n];
    }
#pragma unroll
    for (int t = 0; t < 4; ++t)
      acc[t] = wmma_f32_16x16x4(a, b[t], acc[t]);
  }

  // C/D layout: lane holds col n16, rows v + 8*(lane>>4)
  const int mhalf = (lane >> 4) * 8;
#pragma unroll
  for (int t = 0; t < 4; ++t) {
    const int col = colbase + t * 16 + n16;
    const float bb = addb ? bnode[col] : 0.0f;
#pragma unroll
    for (int v = 0; v < 8; ++v) {
      const int mm = rowTile * 16 + v + mhalf;
      out[(long)mm * outW + col] = acc[t][v] + bb;
    }
  }
}

// ---------------------------------------------------------------------------
// Kernel 3: edge GEMM (efeats @ W_fij) via WMMA, fused with:
//   gather f_ni[src] + f_nj[dst] + bias, leaky_relu(0.01), store f_out,
//   per-head attention logits (16-lane shfl reduction), atomic segment-max.
// One wave per 16-edge tile; 4 heads = 4 col tiles of 16 (FE == 16).
// ---------------------------------------------------------------------------
__global__ __launch_bounds__(32)
void egat_edge(const float* __restrict__ ef,
               const float* __restrict__ Wfij,
               const int*   __restrict__ src,
               const int*   __restrict__ dst,
               const float* __restrict__ f_ni,
               const float* __restrict__ f_nj,
               const float* __restrict__ bias,
               const float* __restrict__ attn,
               float* __restrict__ f_out,     // [E, 64] region of d_out
               float* __restrict__ e_logit,   // [E, 4]
               float* __restrict__ seg_max) { // [N, 4]
  __shared__ int s_src[16];
  __shared__ int s_dst[16];

  const int lane  = threadIdx.x;
  const int eTile = blockIdx.x;
  const int n16   = lane & 15;
  const int kboff = (lane >> 4) * 2;
  const int erow  = eTile * 16 + n16;

  v8f acc[4];
#pragma unroll
  for (int t = 0; t < 4; ++t)
#pragma unroll
    for (int v = 0; v < 8; ++v) acc[t][v] = 0.0f;

  const float* arow = ef + (long)erow * IN_E;
  for (int k = 0; k < IN_E; k += 4) {
    v2f a;
    a.x = arow[k + kboff];
    a.y = arow[k + kboff + 1];
    v2f b[4];
#pragma unroll
    for (int t = 0; t < 4; ++t) {
      const int n = t * 16 + n16;
      b[t].x = Wfij[(long)(k + kboff)     * HFE + n];
      b[t].y = Wfij[(long)(k + kboff + 1) * HFE + n];
    }
#pragma unroll
    for (int t = 0; t < 4; ++t)
      acc[t] = wmma_f32_16x16x4(a, b[t], acc[t]);
  }

  if (lane < 16) {
    s_src[lane] = src[eTile * 16 + lane];
    s_dst[lane] = dst[eTile * 16 + lane];
  }
  __syncthreads();

  const int mhalf = (lane >> 4) * 8;
#pragma unroll
  for (int t = 0; t < 4; ++t) {
    const float at = attn[t * 16 + n16];
    const float bz = bias[t * 16 + n16];
#pragma unroll
    for (int v = 0; v < 8; ++v) {
      const int m = v + mhalf;            // edge within tile
      const int e = eTile * 16 + m;
      const int s = s_src[m];
      const int d = s_dst[m];
      float val = acc[t][v]
                + f_ni[(long)s * HFE + t * 16 + n16]
                + f_nj[(long)d * HFE + t * 16 + n16]
                + bz;
      val = (val > 0.0f) ? val : 0.01f * val;   // leaky_relu
      f_out[(long)e * HFE + t * 16 + n16] = val;

      // logit = sum over FE (16 lanes of this half) of val*attn
      float p = val * at;
      p += __shfl_xor(p, 1, 32);
      p += __shfl_xor(p, 2, 32);
      p += __shfl_xor(p, 4, 32);
      p += __shfl_xor(p, 8, 32);
      if (n16 == 0) {
        e_logit[(long)e * HH + t] = p;
        atomicMaxFloat(&seg_max[(long)d * HH + t], p);
      }
    }
  }
}

// ---------------------------------------------------------------------------
// Kernel 4: e_exp = exp(logit - seg_max[dst]); seg_sum += e_exp
// (e_exp overwrites e_logit in place; kernel 3 rewrites it each call)
// ---------------------------------------------------------------------------
__global__ void egat_softmax(const int* __restrict__ dst,
                             float* __restrict__ e_logit,
                             const float* __restrict__ seg_max,
                             float* __restrict__ seg_sum) {
  long idx = (long)blockIdx.x * blockDim.x + threadIdx.x;
  if (idx >= (long)EE * HH) return;
  const long e = idx >> 2;
  const int  t = (int)(idx & 3);
  const int  d = dst[e];
  const float x = __expf(e_logit[idx] - seg_max[(long)d * HH + t]);
  e_logit[idx] = x;
  atomicAdd(&seg_sum[(long)d * HH + t], x);
}

// ---------------------------------------------------------------------------
// Kernel 5: weighted scatter-sum of h[src] into h_out[dst]
// One thread per 4 consecutive columns: amortizes src/dst/e_exp/seg_sum loads
// over 4 atomics and reads h[src] as a single b128 load.
// ---------------------------------------------------------------------------
__global__ void egat_scatter(const int* __restrict__ src,
                             const int* __restrict__ dst,
                             const float* __restrict__ h,       // [N,128]
                             const float* __restrict__ e_exp,   // [E,4]
                             const float* __restrict__ seg_sum, // [N,4]
                             float* __restrict__ h_out) {       // [N,128]
  long idx = (long)blockIdx.x * blockDim.x + threadIdx.x;
  if (idx >= (long)EE * (HFN / 4)) return;
  const long e = idx >> 5;            // 32 threads per edge
  const int  q = (int)(idx & 31);
  const int  c = q * 4;               // column base (0..124)
  const int  t = q >> 3;              // head = c / 32
  const int  s = src[e];
  const int  d = dst[e];
  const float a = e_exp[(long)e * HH + t] / seg_sum[(long)d * HH + t];
  const float4 hv = *(const float4*)(h + (long)s * HFN + c);
  float* o = h_out + (long)d * HFN + c;
  atomicAdd(o + 0, hv.x * a);
  atomicAdd(o + 1, hv.y * a);
  atomicAdd(o + 2, hv.z * a);
  atomicAdd(o + 3, hv.w * a);
}

// ---------------------------------------------------------------------------
extern "C" void kernel_launch(void* const* d_in, const int* in_sizes, int n_in,
                              void* d_out, int out_size, void* d_ws, size_t ws_size,
                              hipStream_t stream) {
  const float* nfeats = (const float*)d_in[0];
  const float* efeats = (const float*)d_in[1];
  const int*   src    = (const int*)d_in[2];
  const int*   dst    = (const int*)d_in[3];
  const float* W_node = (const float*)d_in[4];
  const float* b_node = (const float*)d_in[5];
  const float* W_ni   = (const float*)d_in[6];
  const float* W_nj   = (const float*)d_in[7];
  const float* W_fij  = (const float*)d_in[8];
  const float* attn   = (const float*)d_in[9];
  const float* bias   = (const float*)d_in[10];

  float* out   = (float*)d_out;
  float* h_out = out;                                   // [N*H, FN] = N*128
  float* f_out = out + (long)NN * HFN;                  // [E*H, FE] = E*64

  // workspace carve-up (floats)
  float* ws      = (float*)d_ws;
  float* f_ni    = ws;                                  // N*64
  float* f_nj    = f_ni    + (long)NN * HFE;            // N*64
  float* h       = f_nj    + (long)NN * HFE;            // N*128
  float* e_logit = h       + (long)NN * HFN;            // E*4 (becomes e_exp)
  float* seg_max = e_logit + (long)EE * HH;             // N*4
  float* seg_sum = seg_max + (long)NN * HH;             // N*4

  // 1) init accumulators (every call: safe under graph replay)
  {
    long n = (long)NN * HFN;
    egat_init<<<(unsigned)((n + 255) / 256), 256, 0, stream>>>(h_out, seg_max, seg_sum);
  }

  // 2) node projections (WMMA f32 16x16x4)
  {
    dim3 grid(NTILES, 4);
    egat_node_proj<<<grid, 32, 0, stream>>>(nfeats, W_ni, W_nj, W_node, b_node,
                                            f_ni, f_nj, h);
  }

  // 3) edge GEMM + gather + leaky_relu + f_out + logits + segment max
  egat_edge<<<ETILES, 32, 0, stream>>>(efeats, W_fij, src, dst, f_ni, f_nj,
                                       bias, attn, f_out, e_logit, seg_max);

  // 4) exp + segment sum
  {
    long n = (long)EE * HH;
    egat_softmax<<<(unsigned)((n + 255) / 256), 256, 0, stream>>>(dst, e_logit,
                                                                  seg_max, seg_sum);
  }

  // 5) weighted scatter-sum (4 columns per thread)
  {
    long n = (long)EE * (HFN / 4);
    egat_scatter<<<(unsigned)((n + 255) / 256), 256, 0, stream>>>(src, dst, h,
                                                                  e_logit, seg_sum,
                                                                  h_out);
  }
}